// Earl_30554397343894
// MI455X (gfx1250) — compile-verified
//
#include <hip/hip_runtime.h>
#include <math.h>

typedef float v2f __attribute__((ext_vector_type(2)));
typedef float v4f __attribute__((ext_vector_type(4)));
typedef float v8f __attribute__((ext_vector_type(8)));

#define DD 256
#define THREE_D 768

__device__ __forceinline__ float sigmoidf_(float x) { return 1.0f / (1.0f + expf(-x)); }

// ---------------- Kernel A: tiny sequential chain (1 block, 256 threads) ----------------
// sub = tanh(sub_W @ sample_mask + sub_b); out[seed] = sub
// r0  = GRUCell(encoder_output, sub)
// gi  = W_ih @ r0 + b_ih   (broadcast input-path for all edges)
__global__ void small_chain_kernel(
    const float* __restrict__ encoder_output,
    const float* __restrict__ sample_mask,
    const float* __restrict__ W_ih, const float* __restrict__ W_hh,
    const float* __restrict__ b_ih, const float* __restrict__ b_hh,
    const float* __restrict__ sub_W, const float* __restrict__ sub_b,
    const int*   __restrict__ seed,
    float* __restrict__ out, float* __restrict__ ws_gi)
{
    __shared__ float sm[DD], enc[DD], sub[DD], r0[DD], gi0[THREE_D], gh0[THREE_D];
    int t = threadIdx.x;
    sm[t] = sample_mask[t];
    enc[t] = encoder_output[t];
    __syncthreads();

    float acc = sub_b[t];
    const float* wrow = sub_W + t * DD;
    for (int k = 0; k < DD; ++k) acc += wrow[k] * sm[k];
    float subv = tanhf(acc);
    sub[t] = subv;
    out[(long long)seed[0] * DD + t] = subv;
    __syncthreads();

    for (int j = t; j < THREE_D; j += DD) {
        float ai = b_ih[j], ah = b_hh[j];
        const float* wi = W_ih + j * DD;
        const float* wh = W_hh + j * DD;
        for (int k = 0; k < DD; ++k) { ai += wi[k] * enc[k]; ah += wh[k] * sub[k]; }
        gi0[j] = ai; gh0[j] = ah;
    }
    __syncthreads();

    float r = sigmoidf_(gi0[t] + gh0[t]);
    float z = sigmoidf_(gi0[DD + t] + gh0[DD + t]);
    float n = tanhf(gi0[2 * DD + t] + r * gh0[2 * DD + t]);
    r0[t] = (1.0f - z) * n + z * sub[t];
    __syncthreads();

    for (int j = t; j < THREE_D; j += DD) {
        float ai = b_ih[j];
        const float* wi = W_ih + j * DD;
        for (int k = 0; k < DD; ++k) ai += wi[k] * r0[k];
        ws_gi[j] = ai;
    }
}

// ---------------- WMMA fp32 GEMM: C = act(A[M,256] @ W[N,256]^T + bias) ----------------
// One wave per 16x16 output tile; K stepped by 4 via V_WMMA_F32_16X16X4_F32.
// A frag (16x4 f32): lane L row = L&15; VGPR0/1 hold K = k0 + hi {+1}, hi = 2*(L>=16).
// B frag (4x16 f32): same K striping; column = tn*16 + (L&15); B[k][n] = W[n][k].
// C/D (16x16 f32, 8 VGPRs): row = tm*16 + v + 8*(L>=16), col = tn*16 + (L&15).
__global__ void wmma_gemm_bias_kernel(
    const float* __restrict__ A, int Mvalid,
    const float* __restrict__ W,
    const float* __restrict__ bias,
    float* __restrict__ C, int Ncols,
    int Mtiles, int Ntiles, int do_tanh)
{
    int wave = (int)((blockIdx.x * blockDim.x + threadIdx.x) >> 5);
    int lane = threadIdx.x & 31;
    if (wave >= Mtiles * Ntiles) return;
    int tm = wave / Ntiles, tn = wave % Ntiles;
    int laneLo = lane & 15;
    int hi = (lane >> 4) << 1;                  // 0 or 2
    int rowA = tm * 16 + laneLo; if (rowA >= Mvalid) rowA = Mvalid - 1;
    int colB = tn * 16 + laneLo;
    const float* aptr = A + (long long)rowA * 256;
    const float* bptr = W + (long long)colB * 256;

    v8f acc = {0.f, 0.f, 0.f, 0.f, 0.f, 0.f, 0.f, 0.f};
    #pragma unroll 8
    for (int k = 0; k < 256; k += 4) {
        v2f a = *(const v2f*)(aptr + k + hi);
        v2f b = *(const v2f*)(bptr + k + hi);
        acc = __builtin_amdgcn_wmma_f32_16x16x4_f32(false, a, false, b, (short)0, acc,
                                                    false, false);
    }

    float bv = bias[colB];
    int rbase = tm * 16 + ((lane >> 4) << 3);
    #pragma unroll
    for (int v = 0; v < 8; ++v) {
        float val = acc[v] + bv;
        if (do_tanh) val = tanhf(val);
        C[(long long)(rbase + v) * Ncols + colB] = val;
    }
}

// ---------------- Kernel C: per-relation GRU gates ----------------
__global__ void gru_rel_kernel(
    const float* __restrict__ gh,        // [Rpad, 768]
    const float* __restrict__ gi,        // [768]
    const float* __restrict__ rel_table, // [R, 256]
    float* __restrict__ rj, int R)
{
    int idx = blockIdx.x * blockDim.x + threadIdx.x;
    if (idx >= R * DD) return;
    int r = idx >> 8, d = idx & 255;
    const float* g = gh + (long long)r * THREE_D;
    float rr = sigmoidf_(gi[d] + g[d]);
    float zz = sigmoidf_(gi[DD + d] + g[DD + d]);
    float nn = tanhf(gi[2 * DD + d] + rr * g[2 * DD + d]);
    rj[idx] = (1.0f - zz) * nn + zz * rel_table[idx];
}

// ---------------- Kernel E: bandwidth-bound gather/scatter ----------------
// 64 v4f slots per 256-float row; each wave's 32 lanes share one edge, so the
// index loads broadcast-coalesce and the state branch is wave-uniform.
// entity_table gathers (single-use, random over 500 MB) and output stores
// (write-once, 205 MB) go non-temporal so the 1 MB obj_table stays hot in L2.
__global__ void scatter_kernel(
    const float* __restrict__ entity_table,
    const float* __restrict__ obj_table,
    const int* __restrict__ rel_ids,
    const int* __restrict__ tail_ids,
    const int* __restrict__ tails_state,
    const int* __restrict__ origin_ids,
    float* __restrict__ out, int E)
{
    long long idx = (long long)blockIdx.x * blockDim.x + threadIdx.x;
    if (idx >= (long long)E * 64) return;
    int e = (int)(idx >> 6);
    int q = (int)(idx & 63);
    v4f val;
    if (tails_state[e] == 1) {
        const v4f* src = (const v4f*)(entity_table + (long long)origin_ids[e] * DD);
        val = __builtin_nontemporal_load(src + q);      // streaming, no reuse
    } else {
        const v4f* src = (const v4f*)(obj_table + (long long)rel_ids[e] * DD);
        val = src[q];                                   // hot in L2 (1 MB, high reuse)
    }
    v4f* dst = (v4f*)(out + (long long)tail_ids[e] * DD);
    __builtin_nontemporal_store(val, dst + q);          // write-once stream
}

extern "C" void kernel_launch(void* const* d_in, const int* in_sizes, int n_in,
                              void* d_out, int out_size, void* d_ws, size_t ws_size,
                              hipStream_t stream) {
    const float* encoder_output = (const float*)d_in[0];
    const float* sample_mask    = (const float*)d_in[1];
    const float* entity_table   = (const float*)d_in[2];
    const float* rel_table      = (const float*)d_in[3];
    const float* W_ih  = (const float*)d_in[4];
    const float* W_hh  = (const float*)d_in[5];
    const float* b_ih  = (const float*)d_in[6];
    const float* b_hh  = (const float*)d_in[7];
    const float* sub_W = (const float*)d_in[8];
    const float* sub_b = (const float*)d_in[9];
    const float* obj_W = (const float*)d_in[10];
    const float* obj_b = (const float*)d_in[11];
    const int* rel_ids     = (const int*)d_in[12];
    const int* tail_ids    = (const int*)d_in[13];
    const int* tails_state = (const int*)d_in[14];
    const int* origin_ids  = (const int*)d_in[15];
    const int* seed        = (const int*)d_in[16];
    float* out = (float*)d_out;

    int E = in_sizes[12];
    int R = in_sizes[3] / DD;
    int Rpad = ((R + 15) / 16) * 16;

    float* ws     = (float*)d_ws;
    float* ws_gi  = ws;                              // 768
    float* ws_gh  = ws_gi + THREE_D;                 // Rpad*768
    float* ws_rj  = ws_gh + (size_t)Rpad * THREE_D;  // Rpad*256
    float* ws_obj = ws_rj + (size_t)Rpad * DD;       // Rpad*256

    small_chain_kernel<<<1, 256, 0, stream>>>(
        encoder_output, sample_mask, W_ih, W_hh, b_ih, b_hh, sub_W, sub_b,
        seed, out, ws_gi);

    int Mtiles = Rpad / 16;
    {   // gh_table = rel_table @ W_hh^T + b_hh   [Rpad, 768]
        int Ntiles = THREE_D / 16;
        int blocks = (Mtiles * Ntiles + 7) / 8;
        wmma_gemm_bias_kernel<<<blocks, 256, 0, stream>>>(
            rel_table, R, W_hh, b_hh, ws_gh, THREE_D, Mtiles, Ntiles, 0);
    }
    {   // rj_table per relation
        int total = R * DD;
        gru_rel_kernel<<<(total + 255) / 256, 256, 0, stream>>>(
            ws_gh, ws_gi, rel_table, ws_rj, R);
    }
    {   // obj_table = tanh(rj_table @ obj_W^T + obj_b)   [Rpad, 256]
        int Ntiles = DD / 16;
        int blocks = (Mtiles * Ntiles + 7) / 8;
        wmma_gemm_bias_kernel<<<blocks, 256, 0, stream>>>(
            ws_rj, R, obj_W, obj_b, ws_obj, DD, Mtiles, Ntiles, 1);
    }
    {   // final gather/scatter (HBM-bound part, ~310-410 MB traffic -> ~16 us)
        long long total = (long long)E * 64;
        int blocks = (int)((total + 255) / 256);
        scatter_kernel<<<blocks, 256, 0, stream>>>(
            entity_table, ws_obj, rel_ids, tail_ids, tails_state, origin_ids, out, E);
    }
}